// DualGCN_44890998177976
// MI455X (gfx1250) — compile-verified
//
#include <hip/hip_runtime.h>
#include <hip/hip_bf16.h>

typedef __attribute__((ext_vector_type(2))) float v2f;
typedef __attribute__((ext_vector_type(8))) float v8f;

#define H 128  // hidden width (h1 == h2 == 128 in the reference)

// ---------------- degree / norm ----------------

__global__ void k_init_deg(float* __restrict__ deg, int N) {
    int i = blockIdx.x * blockDim.x + threadIdx.x;
    if (i < N) deg[i] = 1.0f;  // self-loop contributes 1
}

__global__ void k_edge_deg(const int* __restrict__ ei, int E, float* __restrict__ deg) {
    int e = blockIdx.x * blockDim.x + threadIdx.x;
    if (e < E) atomicAdd(&deg[ei[E + e]], 1.0f);  // dst = ei[1][e]
}

// inv = rsqrt(deg) in place; s initialized with self-loop term inv^2 * x
__global__ void k_inv_s(const float* __restrict__ x, float* __restrict__ deg_inv,
                        float* __restrict__ s, int N) {
    int i = blockIdx.x * blockDim.x + threadIdx.x;
    if (i < N) {
        float inv = rsqrtf(deg_inv[i]);  // deg >= 1 always
        deg_inv[i] = inv;
        s[i] = inv * inv * x[i];
    }
}

// layer-1 scalar aggregation (rank-1 trick): s[dst] += inv[src]*inv[dst]*x[src]
__global__ void k_edge_s(const int* __restrict__ ei, int E, const float* __restrict__ x,
                         const float* __restrict__ inv, float* __restrict__ s) {
    int e = blockIdx.x * blockDim.x + threadIdx.x;
    if (e < E) {
        int src = ei[e], dst = ei[E + e];
        atomicAdd(&s[dst], inv[src] * inv[dst] * x[src]);
    }
}

// h1 = relu(s * W1 + b1); g1 seeded with layer-2 self-loop term inv^2 * h1
__global__ void k_h1(const float* __restrict__ s, const float* __restrict__ inv,
                     const float* __restrict__ W1, const float* __restrict__ b1,
                     float* __restrict__ h1, float* __restrict__ g1, int N) {
    int i = blockIdx.x * blockDim.x + threadIdx.x;
    if (i < N * H) {
        int n = i >> 7, c = i & (H - 1);
        float v = s[n] * W1[c] + b1[c];
        v = v > 0.f ? v : 0.f;
        h1[i] = v;
        float in = inv[n];
        g1[i] = in * in * v;
    }
}

// layer-2 edge aggregation: g1[dst][:] += inv[src]*inv[dst] * h1[src][:]
// 32 lanes per edge, float4 per lane -> each wave streams one full 512B row.
__global__ void k_edge_vec(const int* __restrict__ ei, int E, const float* __restrict__ inv,
                           const float* __restrict__ h1, float* __restrict__ g1) {
    long long gid = (long long)blockIdx.x * blockDim.x + threadIdx.x;
    int e = (int)(gid >> 5);
    int grp = (int)(gid & 31);
    if (e < E) {
        int src = ei[e], dst = ei[E + e];
        float coef = inv[src] * inv[dst];
        const float4 hv = *(const float4*)(h1 + src * H + grp * 4);
        float* gp = g1 + dst * H + grp * 4;
        atomicAdd(gp + 0, coef * hv.x);
        atomicAdd(gp + 1, coef * hv.y);
        atomicAdd(gp + 2, coef * hv.z);
        atomicAdd(gp + 3, coef * hv.w);
    }
}

// Fused h2 = relu(g1 @ W2 + b2) and global-add-pool, via fp32 WMMA.
// Block = 256 threads = 8 waves; wave w owns columns [16w, 16w+16); block owns 16 rows.
// N must be a multiple of 16 (true for 50000 / 100000); tail kernel covers remainders.
__global__ void k_gemm_pool(const float* __restrict__ g1, const float* __restrict__ W2,
                            const float* __restrict__ b2, float* __restrict__ pooled, int N) {
    const int m0   = blockIdx.x * 16;
    const int n0   = (threadIdx.x >> 5) * 16;
    const int lane = threadIdx.x & 31;
    const int half = lane >> 4;   // K sub-pair select (ISA 16x4 A layout)
    const int l    = lane & 15;   // row (A) / column (B,D) within tile

    const float* arow = g1 + (m0 + l) * H;
    v8f acc = {};
#pragma unroll
    for (int k = 0; k < H; k += 4) {
        v2f a = *(const v2f*)(arow + k + 2 * half);          // A[m0+l][k+2h .. +1]
        v2f b;
        b.x = W2[(k + 2 * half) * H + n0 + l];               // B[k+2h][n0+l]
        b.y = W2[(k + 2 * half + 1) * H + n0 + l];
        acc = __builtin_amdgcn_wmma_f32_16x16x4_f32(
            /*neg_a=*/false, a, /*neg_b=*/false, b,
            /*c_mod=*/(short)0, acc, /*reuse_a=*/false, /*reuse_b=*/false);
    }
    // D VGPR j -> column n0+l regardless of half; bias + relu + column partial sum
    float bias = b2[n0 + l];
    float colsum = 0.f;
#pragma unroll
    for (int j = 0; j < 8; ++j) {
        float v = acc[j] + bias;
        colsum += v > 0.f ? v : 0.f;
    }
    atomicAdd(&pooled[n0 + l], colsum);
}

// scalar fallback for rows beyond the last full 16-row tile (unused for given sizes)
__global__ void k_gemm_tail(const float* __restrict__ g1, const float* __restrict__ W2,
                            const float* __restrict__ b2, float* __restrict__ pooled,
                            int N, int start) {
    int idx = blockIdx.x * blockDim.x + threadIdx.x;
    int rows = N - start;
    if (idx < rows * H) {
        int r = start + idx / H, c = idx % H;
        float acc = b2[c];
        for (int k = 0; k < H; ++k) acc += g1[r * H + k] * W2[k * H + c];
        acc = acc > 0.f ? acc : 0.f;
        atomicAdd(&pooled[c], acc);
    }
}

__global__ void k_zero(float* __restrict__ p, int n) {
    int i = blockIdx.x * blockDim.x + threadIdx.x;
    if (i < n) p[i] = 0.f;
}

// out = [pooled_mol, pooled_prot, phys] . Wfc + bfc
__global__ void k_final(const float* __restrict__ pooled, const float* __restrict__ phys,
                        const float* __restrict__ Wfc, const float* __restrict__ bfc,
                        float* __restrict__ out) {
    __shared__ float red[256];
    int t = threadIdx.x;
    float v = pooled[t] * Wfc[t];
    if (t == 0) v += phys[0] * Wfc[256] + bfc[0];
    red[t] = v;
    __syncthreads();
    for (int sft = 128; sft > 0; sft >>= 1) {
        if (t < sft) red[t] += red[t + sft];
        __syncthreads();
    }
    if (t == 0) out[0] = red[0];
}

// ---------------- host side ----------------

static void run_branch(const float* x, const int* ei, int N, int E,
                       const float* W1, const float* b1, const float* W2, const float* b2,
                       float* deg, float* s, float* h1, float* g1, float* pooled,
                       hipStream_t stream) {
    const int tb = 256;
    k_init_deg<<<(N + tb - 1) / tb, tb, 0, stream>>>(deg, N);
    k_edge_deg<<<(E + tb - 1) / tb, tb, 0, stream>>>(ei, E, deg);
    k_inv_s<<<(N + tb - 1) / tb, tb, 0, stream>>>(x, deg, s, N);
    k_edge_s<<<(E + tb - 1) / tb, tb, 0, stream>>>(ei, E, x, deg, s);
    long long nh = (long long)N * H;
    k_h1<<<(unsigned)((nh + tb - 1) / tb), tb, 0, stream>>>(s, deg, W1, b1, h1, g1, N);
    long long tot = (long long)E * 32;
    k_edge_vec<<<(unsigned)((tot + tb - 1) / tb), tb, 0, stream>>>(ei, E, deg, h1, g1);
    int full = N / 16;
    if (full > 0)
        k_gemm_pool<<<full, 256, 0, stream>>>(g1, W2, b2, pooled, N);
    int rem = N - full * 16;
    if (rem > 0) {
        int work = rem * H;
        k_gemm_tail<<<(work + tb - 1) / tb, tb, 0, stream>>>(g1, W2, b2, pooled, N, full * 16);
    }
}

extern "C" void kernel_launch(void* const* d_in, const int* in_sizes, int n_in,
                              void* d_out, int out_size, void* d_ws, size_t ws_size,
                              hipStream_t stream) {
    const float* mol_x   = (const float*)d_in[0];
    const int*   mol_ei  = (const int*)d_in[1];
    const float* prot_x  = (const float*)d_in[2];
    const int*   prot_ei = (const int*)d_in[3];
    const float* phys    = (const float*)d_in[4];
    const float* W1      = (const float*)d_in[5];
    const float* b1      = (const float*)d_in[6];
    const float* W2      = (const float*)d_in[7];
    const float* b2      = (const float*)d_in[8];
    const float* Wfc     = (const float*)d_in[9];
    const float* bfc     = (const float*)d_in[10];

    const int Nm = in_sizes[0];
    const int Em = in_sizes[1] / 2;
    const int Np = in_sizes[2];
    const int Ep = in_sizes[3] / 2;

    float* ws    = (float*)d_ws;
    float* deg_m = ws;
    float* s_m   = deg_m + Nm;
    float* h1_m  = s_m + Nm;
    float* g1_m  = h1_m + (size_t)Nm * H;
    float* deg_p = g1_m + (size_t)Nm * H;
    float* s_p   = deg_p + Np;
    float* h1_p  = s_p + Np;
    float* g1_p  = h1_p + (size_t)Np * H;
    float* pooled = g1_p + (size_t)Np * H;  // 256 floats: [mol 128 | prot 128]

    k_zero<<<1, 256, 0, stream>>>(pooled, 256);

    run_branch(mol_x, mol_ei, Nm, Em, W1, b1, W2, b2,
               deg_m, s_m, h1_m, g1_m, pooled, stream);
    run_branch(prot_x, prot_ei, Np, Ep, W1, b1, W2, b2,
               deg_p, s_p, h1_p, g1_p, pooled + H, stream);

    k_final<<<1, 256, 0, stream>>>(pooled, phys, Wfc, bfc, (float*)d_out);
}